// NeighborList_76716705841438
// MI455X (gfx1250) — compile-verified
//
#include <hip/hip_runtime.h>
#include <hip/hip_bf16.h>

// Neighbor-pair search, store-bandwidth bound (~201 MB out -> ~8.6 us floor
// at 23.3 TB/s). One block per row i; xyz tail staged into LDS via the
// gfx1250 async global->LDS path (global_load_async_to_lds_b128 + ASYNCcnt).

#define NATOMS_MAX 4096
#define BLOCK 256

// Builtin signature (from hipcc diagnostic): first param is
// 'int __attribute__((vector_size(16))) addrspace(1)*'.
typedef int int4v __attribute__((ext_vector_type(4)));
typedef __attribute__((address_space(1))) int4v* g4p;   // global int4*
typedef __attribute__((address_space(3))) int4v* l4p;   // LDS    int4*

#if defined(__gfx1250__) && __has_builtin(__builtin_amdgcn_global_load_async_to_lds_b128)
#define HAVE_ASYNC_LDS 1
#else
#define HAVE_ASYNC_LDS 0
#endif

__device__ __forceinline__ void wait_async_zero() {
#if defined(__gfx1250__)
#if __has_builtin(__builtin_amdgcn_s_wait_asynccnt)
  __builtin_amdgcn_s_wait_asynccnt(0);
#else
  asm volatile("s_wait_asynccnt 0" ::: "memory");
#endif
#endif
}

__global__ __launch_bounds__(BLOCK) void neighbor_rows_kernel(
    const float* __restrict__ xyz,
    const float* __restrict__ cell,
    int*   __restrict__ out_i,
    int*   __restrict__ out_j,
    float* __restrict__ out_delta,
    float* __restrict__ out_dist,
    int n, float cutoff)
{
  __shared__ __align__(16) float ljs[NATOMS_MAX * 3];

  const int i   = (int)blockIdx.x;
  const int tid = (int)threadIdx.x;

  // Stage xyz[jb .. n) into LDS. jb aligned to 4 atoms so the global byte
  // offset (12*jb) and transfer size are 16-byte aligned/exact.
  const int jb     = (i + 1) & ~3;
  const int nfloat = 3 * (n - jb);     // multiple of 12 (and of 4)
  const int nvec   = nfloat >> 2;      // 16-byte chunks
  const float* gsrc = xyz + 3 * jb;

#if HAVE_ASYNC_LDS
  for (int v = tid; v < nvec; v += BLOCK) {
    __builtin_amdgcn_global_load_async_to_lds_b128(
        (g4p)(gsrc + 4 * v),
        (l4p)(&ljs[4 * v]),
        0 /*imm offset*/, 0 /*cpol*/);
  }
  wait_async_zero();   // each wave waits for its own async copies
  __syncthreads();     // then cross-wave visibility in LDS
#else
  {
    const float4* g4 = reinterpret_cast<const float4*>(gsrc);
    float4*       l4 = reinterpret_cast<float4*>(ljs);
    for (int v = tid; v < nvec; v += BLOCK) l4[v] = g4[v];
  }
  __syncthreads();
#endif

  const int L = n - 1 - i;             // pairs in this row
  if (L <= 0) return;

  // Cell (uniform across all threads; loads broadcast from L2).
  const float c00 = cell[0], c01 = cell[1], c02 = cell[2];
  const float c10 = cell[3], c11 = cell[4], c12 = cell[5];
  const float c20 = cell[6], c21 = cell[7], c22 = cell[8];
  const bool pbc = !(c00 == 0.f && c01 == 0.f && c02 == 0.f &&
                     c10 == 0.f && c11 == 0.f && c12 == 0.f &&
                     c20 == 0.f && c21 == 0.f && c22 == 0.f);

  // inv(cell) via adjugate (only meaningful when pbc).
  const float A =  (c11 * c22 - c12 * c21);
  const float B = -(c10 * c22 - c12 * c20);
  const float C =  (c10 * c21 - c11 * c20);
  const float det  = c00 * A + c01 * B + c02 * C;
  const float rdet = pbc ? (1.0f / det) : 0.0f;
  const float i00 = A * rdet;
  const float i10 = B * rdet;
  const float i20 = C * rdet;
  const float i01 = -(c01 * c22 - c02 * c21) * rdet;
  const float i11 =  (c00 * c22 - c02 * c20) * rdet;
  const float i21 = -(c00 * c21 - c01 * c20) * rdet;
  const float i02 =  (c01 * c12 - c02 * c11) * rdet;
  const float i12 = -(c00 * c12 - c02 * c10) * rdet;
  const float i22 =  (c00 * c11 - c01 * c10) * rdet;

  const float xi0 = xyz[3 * i + 0];
  const float xi1 = xyz[3 * i + 1];
  const float xi2 = xyz[3 * i + 2];

  // Flat offset of row i in the i<j triangular pair list.
  const int rowoff = i * (n - 1) - (i * (i - 1)) / 2;

  for (int t = tid; t < L; t += BLOCK) {
    const int j  = i + 1 + t;
    const int lj = 3 * (j - jb);
    float d0 = xi0 - ljs[lj + 0];
    float d1 = xi1 - ljs[lj + 1];
    float d2 = xi2 - ljs[lj + 2];

    if (pbc) {  // uniform branch
      const float f0 = d0 * i00 + d1 * i10 + d2 * i20;
      const float f1 = d0 * i01 + d1 * i11 + d2 * i21;
      const float f2 = d0 * i02 + d1 * i12 + d2 * i22;
      const float r0 = rintf(f0);   // RNE, matches jnp.round
      const float r1 = rintf(f1);
      const float r2 = rintf(f2);
      d0 -= r0 * c00 + r1 * c10 + r2 * c20;
      d1 -= r0 * c01 + r1 * c11 + r2 * c21;
      d2 -= r0 * c02 + r1 * c12 + r2 * c22;
    }

    const float dist = sqrtf(d0 * d0 + d1 * d1 + d2 * d2);
    const bool  m    = dist <= cutoff;
    const int   p    = rowoff + t;

    out_i[p] = m ? i : -1;
    out_j[p] = m ? j : -1;
    out_delta[3 * p + 0] = m ? d0 : 0.0f;   // adjacent dwords -> b96 store
    out_delta[3 * p + 1] = m ? d1 : 0.0f;
    out_delta[3 * p + 2] = m ? d2 : 0.0f;
    out_dist[p] = m ? dist : 0.0f;
  }
}

extern "C" void kernel_launch(void* const* d_in, const int* in_sizes, int n_in,
                              void* d_out, int out_size, void* d_ws, size_t ws_size,
                              hipStream_t stream) {
  const float* xyz  = (const float*)d_in[0];   // [n,3] f32
  const float* cell = (const float*)d_in[1];   // [3,3] f32

  const int n = in_sizes[0] / 3;               // 4096
  const long long P = (long long)n * (n - 1) / 2;

  float* base      = (float*)d_out;            // 6P dwords total
  int*   out_i     = (int*)base;               // [0,   P)  int32
  int*   out_j     = (int*)(base + P);         // [P,  2P)  int32
  float* out_delta = base + 2 * P;             // [2P, 5P)  f32 [P,3]
  float* out_dist  = base + 5 * P;             // [5P, 6P)  f32

  neighbor_rows_kernel<<<dim3((unsigned)n), dim3(BLOCK), 0, stream>>>(
      xyz, cell, out_i, out_j, out_delta, out_dist, n, 5.0f);
}